// ModelNew_49460843381584
// MI455X (gfx1250) — compile-verified
//
#include <hip/hip_runtime.h>
#include <hip/hip_bf16.h>

// Problem constants (match reference.py)
#define BATCH   16384
#define IN_F    4096
#define OUT_F   4096
#define NSPLIT  16          // row splits for the W column-reduction

typedef __attribute__((ext_vector_type(2))) float v2f;
typedef __attribute__((ext_vector_type(8))) float v8f;

// -----------------------------------------------------------------------------
// Kernel 1: column-sum of W via V_WMMA_F32_16X16X4_F32.
// A = all-ones 16x4  =>  D[m,n] = sum_k B[k,n] + C[m,n]  (pure column reduce).
// Each single-wave workgroup owns a 16-column tile and a 256-row split;
// every WMMA consumes 4 rows x 16 cols of W through the B operand.
// Result is layout-robust: any K-row permutation in B sums identically per column.
// -----------------------------------------------------------------------------
__global__ __launch_bounds__(32) void colsum_wmma_kernel(
    const float* __restrict__ W,      // (OUT_F, IN_F) row-major
    float* __restrict__ partial,      // (NSPLIT, IN_F)
    int rows_per_split)
{
    const int col_base = blockIdx.x * 16;               // column tile
    const int row_base = blockIdx.y * rows_per_split;   // row split
    const int lane = threadIdx.x;                       // wave32
    const int hi   = lane >> 4;                         // 0: lanes 0-15, 1: lanes 16-31
    const int nl   = lane & 15;                         // N index within tile

    const float* wp = W + (size_t)row_base * IN_F + col_base + nl;

    v2f a; a.x = 1.0f; a.y = 1.0f;                      // all-ones A (16x4)
    v8f c = {};                                         // accumulator (16x16 f32)

    #pragma unroll 4
    for (int r = 0; r < rows_per_split; r += 4) {
        const float* p = wp + (size_t)r * IN_F;
        v2f b;
        // B tile 4x16: VGPR0 <- rows {0,1} split by lane-half, VGPR1 <- rows {2,3}
        b.x = p[(size_t)hi * IN_F];
        b.y = p[(size_t)(2 + hi) * IN_F];
        c = __builtin_amdgcn_wmma_f32_16x16x4_f32(
                /*neg_a=*/false, a, /*neg_b=*/false, b,
                /*c_mod=*/(short)0, c, /*reuse_a=*/false, /*reuse_b=*/false);
    }

    // D row M=0 lives in c[0] on lanes 0-15 (N = lane).
    if (hi == 0) {
        partial[(size_t)blockIdx.y * IN_F + col_base + nl] = c[0];
    }
}

// -----------------------------------------------------------------------------
// Kernel 2: fold the NSPLIT partial column-sums into the final colsum vector.
// -----------------------------------------------------------------------------
__global__ __launch_bounds__(256) void reduce_partials_kernel(
    const float* __restrict__ partial,  // (NSPLIT, IN_F)
    float* __restrict__ colsum)         // (IN_F)
{
    const int i = blockIdx.x * blockDim.x + threadIdx.x;
    if (i < IN_F) {
        float s = 0.0f;
        #pragma unroll
        for (int j = 0; j < NSPLIT; ++j) s += partial[(size_t)j * IN_F + i];
        colsum[i] = s;
    }
}

// -----------------------------------------------------------------------------
// Kernel 3: out[b] = dot(x[b,:], colsum).  This is the 256 MB stream that sets
// the runtime, so it gets the bandwidth-optimal shape: one wave per row,
// float4 (global_load_b128) fully-coalesced loads, wave32 shuffle reduction.
// colsum (16 KB) stays resident in cache across all waves.
// -----------------------------------------------------------------------------
__global__ __launch_bounds__(256) void gemv_dot_kernel(
    const float* __restrict__ x,       // (BATCH, IN_F)
    const float* __restrict__ colsum,  // (IN_F)
    float* __restrict__ out)           // (BATCH)
{
    const int gtid = blockIdx.x * blockDim.x + threadIdx.x;
    const int row  = gtid >> 5;        // one wave32 per batch row
    const int lane = gtid & 31;
    if (row >= BATCH) return;

    const float* xp = x + (size_t)row * IN_F;

    float s = 0.0f;
    #pragma unroll 4
    for (int i = lane * 4; i < IN_F; i += 32 * 4) {
        float4 v = *(const float4*)(xp + i);
        float4 w = *(const float4*)(colsum + i);
        s = fmaf(v.x, w.x, s);
        s = fmaf(v.y, w.y, s);
        s = fmaf(v.z, w.z, s);
        s = fmaf(v.w, w.w, s);
    }

    // wave32 butterfly reduction
    #pragma unroll
    for (int off = 16; off > 0; off >>= 1)
        s += __shfl_xor(s, off, 32);

    if (lane == 0) out[row] = s;
}

// -----------------------------------------------------------------------------
// Launch: zero external state, no atomics, d_ws fully rewritten each call.
// Workspace: partial[NSPLIT][IN_F] then colsum[IN_F]  => (16+1)*4096*4 = 272 KB.
// -----------------------------------------------------------------------------
extern "C" void kernel_launch(void* const* d_in, const int* in_sizes, int n_in,
                              void* d_out, int out_size, void* d_ws, size_t ws_size,
                              hipStream_t stream) {
    const float* x = (const float*)d_in[0];   // (BATCH, IN_F) float32
    const float* W = (const float*)d_in[1];   // (OUT_F, IN_F) float32
    float* out = (float*)d_out;               // (BATCH, 1) float32

    float* partial = (float*)d_ws;                      // NSPLIT * IN_F floats
    float* colsum  = partial + (size_t)NSPLIT * IN_F;   // IN_F floats (16B aligned)

    // Pass 1: W column reduction through the WMMA pipe.
    dim3 g1(IN_F / 16, NSPLIT);
    colsum_wmma_kernel<<<g1, 32, 0, stream>>>(W, partial, OUT_F / NSPLIT);

    // Pass 2: fold partials.
    reduce_partials_kernel<<<(IN_F + 255) / 256, 256, 0, stream>>>(partial, colsum);

    // Pass 3: bandwidth-bound GEMV over x.
    const int threads = 256;                   // 8 waves per block
    const int total   = BATCH * 32;            // one wave per row
    gemv_dot_kernel<<<(total + threads - 1) / threads, threads, 0, stream>>>(x, colsum, out);
}